// StreamNet_K3_S1_8186207666437
// MI455X (gfx1250) — compile-verified
//
#include <hip/hip_runtime.h>

// CDNA5 / gfx1250: wave32, WMMA f32 16x16x4 (exact fp32 matrix math).
typedef __attribute__((ext_vector_type(2))) float v2f;
typedef __attribute__((ext_vector_type(8))) float v8f;

#define BSZ   8
#define CCH   32
#define PSZ   256
#define TW    16            // output tile cols per workgroup
#define TH    16            // output tile rows per workgroup
#define VR    (TH + 2)      // 18 virtual rows (2-row top halo)
#define VC    (TW + 2)      // 18 virtual cols (2-col left halo)
#define NPIX  (VR * VC)     // 324
#define CHS   34            // channel stride in LDS (padded 32->34: bank-conflict-free A loads)

__global__ __launch_bounds__(256)
void streamnet_conv3x3_wmma_f32(const float* __restrict__ x,
                                const float* __restrict__ rbuf,
                                const float* __restrict__ bbuf,
                                const float* __restrict__ Wg,
                                const float* __restrict__ bias,
                                float* __restrict__ out)
{
    __shared__ float tile[NPIX * CHS];   // 324*34*4 = 44,064 B

    const int b   = blockIdx.z;
    const int OY  = blockIdx.y * TH;
    const int OX  = blockIdx.x * TW;
    const int tid = threadIdx.x;

    // ---- stage halo-stitched input tile into LDS ([pixel][channel], padded) ----
    for (int i = tid; i < CCH * NPIX; i += 256) {
        const int ch  = i / NPIX;
        const int rem = i - ch * NPIX;
        const int vr  = rem / VC;
        const int vc  = rem - vr * VC;
        const int vy  = OY + vr - 2;     // virtual row in [-2, 256)
        const int vx  = OX + vc - 2;     // virtual col in [-2, 256)
        float v;
        if (vy < 0) {
            // bottom-halo from top neighbor: [B,C,2,P+2], spans full 258 width
            v = bbuf[((b * CCH + ch) * 2 + (vy + 2)) * (PSZ + 2) + (vx + 2)];
        } else if (vx < 0) {
            // right-halo from left neighbor: [B,C,P,2]
            v = rbuf[((b * CCH + ch) * PSZ + vy) * 2 + (vx + 2)];
        } else if (vx == PSZ - 1) {
            // output_bypassing: x's last column zeroed
            v = 0.0f;
        } else {
            v = x[((b * CCH + ch) * PSZ + vy) * PSZ + vx];
        }
        tile[(vr * VC + vc) * CHS + ch] = v;
    }
    __syncthreads();

    // ---- per-wave tile assignment (8 waves: 2 N-groups x 4 row-quads) ----
    const int lane   = tid & 31;
    const int wv     = tid >> 5;
    const int nGrp   = wv & 1;           // which 16 output channels
    const int tyBase = (wv >> 1) * 4;    // 4 output rows per wave
    const int lHalf  = lane >> 4;        // 0: K pair (k,k+1), 1: (k+2,k+3)
    const int lIdx   = lane & 15;        // M (pixel/col) for A, N (channel) for B/D
    const int ch     = nGrp * 16 + lIdx; // output channel this lane produces

    const float bv = bias[ch];
    v8f acc[4];
#pragma unroll
    for (int j = 0; j < 4; ++j)
        acc[j] = (v8f){bv, bv, bv, bv, bv, bv, bv, bv};

    // ---- 9 taps x (K=32 as 8 k-steps of 4) accumulated GEMMs ----
    for (int tap = 0; tap < 9; ++tap) {
        const int ky = tap / 3;
        const int kx = tap - ky * 3;

        // B[k][n] = W[c_out=ch][c_in=k][ky][kx]; hoisted per tap, reused over 4 M-tiles
        v2f Breg[8];
#pragma unroll
        for (int kk = 0; kk < 8; ++kk) {
            const int k0 = kk * 4 + 2 * lHalf;
            Breg[kk].x = Wg[(ch * CCH + k0)     * 9 + tap];
            Breg[kk].y = Wg[(ch * CCH + k0 + 1) * 9 + tap];
        }

#pragma unroll
        for (int j = 0; j < 4; ++j) {
            const int ty = tyBase + j;
            // A[m=lIdx][k]: lane reads pixel (ty+ky, lIdx+kx), channels k0,k0+1
            const float* ap =
                &tile[((ty + ky) * VC + (lIdx + kx)) * CHS + 2 * lHalf];
#pragma unroll
            for (int kk = 0; kk < 8; ++kk) {
                v2f a = *(const v2f*)(ap + kk * 4);   // ds_load_b64, conflict-free
                acc[j] = __builtin_amdgcn_wmma_f32_16x16x4_f32(
                    false, a, false, Breg[kk], (short)0, acc[j], false, false);
            }
        }
    }

    // ---- writeback: D VGPR g -> (col = g + 8*lHalf, channel = ch) ----
#pragma unroll
    for (int j = 0; j < 4; ++j) {
        const int oy = OY + tyBase + j;
        float* op = &out[((b * CCH + ch) * PSZ + oy) * PSZ + OX + 8 * lHalf];
#pragma unroll
        for (int g = 0; g < 8; ++g)
            op[g] = acc[j][g];
    }
}

extern "C" void kernel_launch(void* const* d_in, const int* in_sizes, int n_in,
                              void* d_out, int out_size, void* d_ws, size_t ws_size,
                              hipStream_t stream) {
    (void)in_sizes; (void)n_in; (void)d_ws; (void)ws_size; (void)out_size;
    const float* x    = (const float*)d_in[0];
    const float* rbuf = (const float*)d_in[1];
    const float* bbuf = (const float*)d_in[2];
    const float* Wg   = (const float*)d_in[3];
    const float* bias = (const float*)d_in[4];
    float* out = (float*)d_out;

    dim3 grid(PSZ / TW, PSZ / TH, BSZ);   // (16, 16, 8) = 2048 workgroups
    streamnet_conv3x3_wmma_f32<<<grid, 256, 0, stream>>>(x, rbuf, bbuf, Wg, bias, out);
}